// FinalNet_49065706390364
// MI455X (gfx1250) — compile-verified
//
#include <hip/hip_runtime.h>

#define DEV __device__ __forceinline__

static constexpr int Bsz = 4096;   // batch
static constexpr int Dm  = 64;     // embed dim
static constexpr int Hh  = 50;     // history length
static constexpr int Kk  = 20;     // top-k (0.4 * 50) for both branches

typedef __attribute__((ext_vector_type(2))) float v2f;
typedef __attribute__((ext_vector_type(8))) float v8f;

#if __has_builtin(__builtin_amdgcn_wmma_f32_16x16x4_f32)
#define USE_WMMA_F32 1
#else
#define USE_WMMA_F32 0
#endif

// ---------------- wave32 reductions ----------------
DEV float wave_sum(float x){
  x += __shfl_xor(x, 1, 32);
  x += __shfl_xor(x, 2, 32);
  x += __shfl_xor(x, 4, 32);
  x += __shfl_xor(x, 8, 32);
  x += __shfl_xor(x, 16, 32);
  return x;
}
DEV float wave_max(float x){
  x = fmaxf(x, __shfl_xor(x, 1, 32));
  x = fmaxf(x, __shfl_xor(x, 2, 32));
  x = fmaxf(x, __shfl_xor(x, 4, 32));
  x = fmaxf(x, __shfl_xor(x, 8, 32));
  x = fmaxf(x, __shfl_xor(x, 16, 32));
  return x;
}

// ---------------- JAX Threefry-2x32 Gumbel noise ----------------
DEV unsigned rotl32(unsigned x, int r){ return (x << r) | (x >> (32 - r)); }

DEV float jax_gumbel(unsigned keylo, unsigned idx){
  // replicates -log(-log(u)), u = jax.random.uniform(key,(B,H),1e-10,1.0)
  const unsigned n    = (unsigned)(Bsz * Hh);   // 204800 (even)
  const unsigned half = n >> 1;
  const bool hi = idx >= half;
  unsigned x0 = hi ? (idx - half) : idx;
  unsigned x1 = x0 + half;
  const unsigned k0 = 0u, k1 = keylo;
  const unsigned k2 = 0x1BD11BDAu ^ k0 ^ k1;
  x0 += k0; x1 += k1;
#define TF4(a,b,c,d) \
  x0 += x1; x1 = rotl32(x1,a); x1 ^= x0; \
  x0 += x1; x1 = rotl32(x1,b); x1 ^= x0; \
  x0 += x1; x1 = rotl32(x1,c); x1 ^= x0; \
  x0 += x1; x1 = rotl32(x1,d); x1 ^= x0;
  TF4(13,15,26,6)  x0 += k1; x1 += k2 + 1u;
  TF4(17,29,16,24) x0 += k2; x1 += k0 + 2u;
  TF4(13,15,26,6)  x0 += k0; x1 += k1 + 3u;
  TF4(17,29,16,24) x0 += k1; x1 += k2 + 4u;
  TF4(13,15,26,6)  x0 += k2; x1 += k0 + 5u;
#undef TF4
  const unsigned bits = hi ? x1 : x0;
  float f = __uint_as_float((bits >> 9) | 0x3f800000u) - 1.0f;
  float u = f * (1.0f - 1e-10f) + 1e-10f;
  u = fmaxf(u, 1e-10f);
  return -logf(-logf(u));
}

// ---------------- wave-parallel top-20 over 64 padded slots ----------------
DEV void topk20_wave(const float* keyv, int lane, int* sel){
  float v0 = keyv[lane], v1 = keyv[lane + 32];
  for (int k = 0; k < Kk; ++k){
    float bv; int bi;
    if (v1 > v0){ bv = v1; bi = lane + 32; } else { bv = v0; bi = lane; }
    for (int off = 16; off > 0; off >>= 1){
      const float ov = __shfl_xor(bv, off, 32);
      const int   oi = __shfl_xor(bi, off, 32);
      if (ov > bv || (ov == bv && oi < bi)){ bv = ov; bi = oi; }
    }
    if (lane == 0) sel[k] = bi;
    if (bi == lane)      v0 = -3.0e38f;
    if (bi == lane + 32) v1 = -3.0e38f;
  }
}

// -------- 16x16 output tile, f32 WMMA over K (A row-major ldA, B row-major ldB) --------
DEV v8f wmma_f32_block(const float* A0, int ldA, const float* B0, int ldB,
                       int Ktot, int lanelo, int lanehi){
#if USE_WMMA_F32
  v8f acc = {};
  const int steps = Ktot >> 2;
  for (int kk = 0; kk < steps; ++kk){
    const int kb = (kk << 2) + (lanehi << 1);       // A/B frag: K = kb + v
    v2f a, b;
    a.x = A0[lanelo * ldA + kb];
    a.y = A0[lanelo * ldA + kb + 1];
    b.x = B0[kb * ldB + lanelo];
    b.y = B0[(kb + 1) * ldB + lanelo];
    acc = __builtin_amdgcn_wmma_f32_16x16x4_f32(false, a, false, b,
                                                (short)0, acc, false, false);
  }
  return acc;
#else
  v8f acc = {};
  for (int v = 0; v < 8; ++v){
    const int row = v + (lanehi << 3);
    float s = 0.f;
    for (int k = 0; k < Ktot; ++k) s += A0[row * ldA + k] * B0[k * ldB + lanelo];
    acc[v] = s;
  }
  return acc;
#endif
}

// ================= kernel 1: query = concat(self,target) @ W_lin + b =================
__global__ __launch_bounds__(256)
void k_query(const int* __restrict__ user, const int* __restrict__ cand,
             const float* __restrict__ user_emb, const float* __restrict__ item_emb,
             const float* __restrict__ W_lin, const float* __restrict__ b_lin,
             float* __restrict__ queryW){
  __shared__ float Alds[32 * 128];
  __shared__ float Wlds[128 * 64];
  __shared__ float blds[64];
  const int t = threadIdx.x, wave = t >> 5, lane = t & 31;
  const int lanelo = lane & 15, lanehi = lane >> 4;
  const int gb0 = blockIdx.x * 32;
  for (int i = t; i < 128 * 64; i += 256) Wlds[i] = W_lin[i];
  if (t < 64) blds[t] = b_lin[t];
  for (int r = wave; r < 32; r += 8){                 // 256B-coalesced gathers
    const int gb = gb0 + r;
    const float* src = (lane < 16)
        ? (user_emb + (size_t)user[gb] * Dm + lane * 4)
        : (item_emb + (size_t)cand[gb] * Dm + (lane - 16) * 4);
    const float4 vls = *(const float4*)src;
    const int col = (lane < 16) ? lane * 4 : 64 + (lane - 16) * 4;
    *(float4*)&Alds[r * 128 + col] = vls;
  }
  __syncthreads();
  const int mt = wave >> 2, nt = wave & 3;            // 8 jobs, 1 per wave
  const v8f acc = wmma_f32_block(&Alds[mt * 16 * 128], 128, &Wlds[nt * 16], 64,
                                 128, lanelo, lanehi);
  const int n = nt * 16 + lanelo;
  const float bias = blds[n];
  for (int v = 0; v < 8; ++v){
    const int row = mt * 16 + v + (lanehi << 3);
    queryW[(size_t)(gb0 + row) * Dm + n] = acc[v] + bias;
  }
}

// ================= kernel 2: history branch -> agg1 =================
__global__ __launch_bounds__(256)
void k_hist(const int* __restrict__ hist_idx, const int* __restrict__ hist_r,
            const float* __restrict__ item_emb, const float* __restrict__ r2e,
            const float* __restrict__ ratt_g, const float* __restrict__ queryW,
            float* __restrict__ agg1){
  __shared__ float euv[50 * 64];
  __shared__ float r2eL[7 * 64];
  __shared__ float qL[64];
  __shared__ float rattL[128];
  __shared__ float keyv[64];
  __shared__ int   sel[Kk];
  __shared__ float sc[Kk];
  __shared__ float prob[Kk];
  const int b = blockIdx.x;
  const int t = threadIdx.x, wave = t >> 5, lane = t & 31;
  if (t < 64)  qL[t]    = queryW[(size_t)b * Dm + t];
  if (t < 128) rattL[t] = ratt_g[t];
  if (t < 64)  keyv[t]  = -3.0e38f;
  for (int i = t; i < 7 * 64; i += 256) r2eL[i] = r2e[i];
  for (int r = wave; r < Hh; r += 8){
    const int it = hist_idx[b * Hh + r];
    const float2 e = *(const float2*)(item_emb + (size_t)it * Dm + lane * 2);
    euv[r * 64 + lane * 2]     = e.x;
    euv[r * 64 + lane * 2 + 1] = e.y;
  }
  __syncthreads();
  // selection key = -||q-e|| + gumbel  (log_softmax shift drops out of top-k)
  for (int r = wave; r < Hh; r += 8){
    const float dx = qL[lane * 2]     - euv[r * 64 + lane * 2];
    const float dy = qL[lane * 2 + 1] - euv[r * 64 + lane * 2 + 1];
    const float s  = wave_sum(dx * dx + dy * dy);
    if (lane == 0) keyv[r] = -sqrtf(s) + jax_gumbel(42u, (unsigned)(b * Hh + r));
  }
  __syncthreads();
  if (wave == 0) topk20_wave(keyv, lane, sel);
  __syncthreads();
  for (int k = wave; k < Kk; k += 8){
    const int s  = sel[k];
    const int rr = hist_r[b * Hh + s];
    const float2 e  = *(const float2*)&euv[s * 64 + lane * 2];
    const float2 re = *(const float2*)&r2eL[rr * 64 + lane * 2];
    float p = e.x * rattL[lane * 2] + e.y * rattL[lane * 2 + 1]
            + re.x * rattL[64 + lane * 2] + re.y * rattL[64 + lane * 2 + 1];
    p = wave_sum(p);
    if (lane == 0) sc[k] = p;
  }
  __syncthreads();
  if (wave == 0){
    const float x = (lane < Kk) ? sc[lane] : -3.0e38f;
    const float m = wave_max(x);
    const float e = (lane < Kk) ? __expf(x - m) : 0.f;
    const float ss = wave_sum(e);
    if (lane < Kk) prob[lane] = e / ss;
  }
  __syncthreads();
  if (t < 64){
    float a = 0.f;
    for (int k = 0; k < Kk; ++k) a += prob[k] * euv[sel[k] * 64 + t];
    agg1[(size_t)b * Dm + t] = a;
  }
}

// ========= kernel 3: friend branch (fused WMMA transform in LDS) -> agg2 =========
__global__ __launch_bounds__(256)
void k_friend(const int* __restrict__ user, const int* __restrict__ friend_idx,
              const float* __restrict__ user_emb, const float* __restrict__ transform,
              const float* __restrict__ W_att1, const float* __restrict__ b_att1,
              const float* __restrict__ queryW, float* __restrict__ agg2){
  __shared__ float en[64 * 64];   // raw neighbor embeddings (padded to 64 rows)
  __shared__ float Tl[64 * 64];   // transform_user
  __shared__ float et[64 * 64];   // transformed neighbors
  __shared__ float qL[64];
  __shared__ float sfL[64];
  __shared__ float wa1[128];
  __shared__ float keyv[64];
  __shared__ int   sel[Kk];
  __shared__ float sc[Kk];
  __shared__ float prob[Kk];
  const int b = blockIdx.x;
  const int t = threadIdx.x, wave = t >> 5, lane = t & 31;
  const int lanelo = lane & 15, lanehi = lane >> 4;
  if (t < 64)  qL[t]   = queryW[(size_t)b * Dm + t];
  if (t < 64)  sfL[t]  = user_emb[(size_t)user[b] * Dm + t];
  if (t < 128) wa1[t]  = W_att1[t];
  if (t < 64)  keyv[t] = -3.0e38f;
  for (int i = t; i < 64 * 64; i += 256) Tl[i] = transform[i];
  for (int r = wave; r < 64; r += 8){
    if (r < Hh){
      const int f = friend_idx[b * Hh + r];
      const float2 e = *(const float2*)(user_emb + (size_t)f * Dm + lane * 2);
      en[r * 64 + lane * 2] = e.x; en[r * 64 + lane * 2 + 1] = e.y;
    } else {
      en[r * 64 + lane * 2] = 0.f; en[r * 64 + lane * 2 + 1] = 0.f;
    }
  }
  __syncthreads();
  // et = en @ transform : 4x4 tiles of 16x16, K=64, all in LDS
  for (int job = wave; job < 16; job += 8){
    const int mt = job >> 2, nt = job & 3;
    const v8f acc = wmma_f32_block(&en[mt * 16 * 64], 64, &Tl[nt * 16], 64,
                                   64, lanelo, lanehi);
    for (int v = 0; v < 8; ++v)
      et[(mt * 16 + v + (lanehi << 3)) * 64 + nt * 16 + lanelo] = acc[v];
  }
  __syncthreads();
  for (int r = wave; r < Hh; r += 8){
    const float dx = qL[lane * 2]     - et[r * 64 + lane * 2];
    const float dy = qL[lane * 2 + 1] - et[r * 64 + lane * 2 + 1];
    const float s  = wave_sum(dx * dx + dy * dy);
    if (lane == 0) keyv[r] = -sqrtf(s) + jax_gumbel(43u, (unsigned)(b * Hh + r));
  }
  __syncthreads();
  if (wave == 0) topk20_wave(keyv, lane, sel);
  __syncthreads();
  const float ba1 = b_att1[0];
  for (int k = wave; k < Kk; k += 8){
    const int s = sel[k];
    float p = sfL[lane * 2] * wa1[lane * 2] + sfL[lane * 2 + 1] * wa1[lane * 2 + 1]
            + en[s * 64 + lane * 2] * wa1[64 + lane * 2]
            + en[s * 64 + lane * 2 + 1] * wa1[64 + lane * 2 + 1];
    p = wave_sum(p);
    if (lane == 0) sc[k] = fmaxf(p + ba1, 0.f);   // relu before softmax
  }
  __syncthreads();
  if (wave == 0){
    const float x = (lane < Kk) ? sc[lane] : -3.0e38f;
    const float m = wave_max(x);
    const float e = (lane < Kk) ? __expf(x - m) : 0.f;
    const float ss = wave_sum(e);
    if (lane < Kk) prob[lane] = e / ss;
  }
  __syncthreads();
  if (t < 64){
    float a = 0.f;
    for (int k = 0; k < Kk; ++k) a += prob[k] * en[sel[k] * 64 + t];
    agg2[(size_t)b * Dm + t] = a;
  }
}

// ========= kernel 4: 2-way fusion attention + final GEMM (relu) =========
__global__ __launch_bounds__(256)
void k_final(const int* __restrict__ user, const float* __restrict__ user_emb,
             const float* __restrict__ agg1, const float* __restrict__ agg2,
             const float* __restrict__ W_att5, const float* __restrict__ b_att5,
             const float* __restrict__ W_lin2, const float* __restrict__ b_lin2,
             float* __restrict__ outp){
  __shared__ float Alds[32 * 128];
  __shared__ float Wlds[128 * 64];
  __shared__ float wa5[128];
  __shared__ float b2L[64];
  const int t = threadIdx.x, wave = t >> 5, lane = t & 31;
  const int lanelo = lane & 15, lanehi = lane >> 4;
  const int gb0 = blockIdx.x * 32;
  for (int i = t; i < 128 * 64; i += 256) Wlds[i] = W_lin2[i];
  if (t < 128) wa5[t] = W_att5[t];
  if (t < 64)  b2L[t] = b_lin2[t];
  __syncthreads();
  const float b5 = b_att5[0];
  for (int r = wave; r < 32; r += 8){
    const int gb = gb0 + r;
    const float2 sf = *(const float2*)(user_emb + (size_t)user[gb] * Dm + lane * 2);
    const float2 a1 = *(const float2*)(agg1 + (size_t)gb * Dm + lane * 2);
    const float2 a2 = *(const float2*)(agg2 + (size_t)gb * Dm + lane * 2);
    const int d0 = lane * 2, d1 = d0 + 1;
    float su = sf.x * wa5[d0]      + sf.y * wa5[d1];
    float s1 = a1.x * wa5[64 + d0] + a1.y * wa5[64 + d1];
    float s2 = a2.x * wa5[64 + d0] + a2.y * wa5[64 + d1];
    su = wave_sum(su); s1 = wave_sum(s1); s2 = wave_sum(s2);
    const float x1 = fmaxf(su + s1 + b5, 0.f);
    const float x2 = fmaxf(su + s2 + b5, 0.f);
    const float m  = fmaxf(x1, x2);
    const float e1 = __expf(x1 - m), e2 = __expf(x2 - m);
    const float w1 = e1 / (e1 + e2), w2 = e2 / (e1 + e2);
    Alds[r * 128 + d0]      = sf.x;
    Alds[r * 128 + d1]      = sf.y;
    Alds[r * 128 + 64 + d0] = w1 * a1.x + w2 * a2.x;
    Alds[r * 128 + 64 + d1] = w1 * a1.y + w2 * a2.y;
  }
  __syncthreads();
  const int mt = wave >> 2, nt = wave & 3;
  const v8f acc = wmma_f32_block(&Alds[mt * 16 * 128], 128, &Wlds[nt * 16], 64,
                                 128, lanelo, lanehi);
  const int n = nt * 16 + lanelo;
  for (int v = 0; v < 8; ++v){
    const int row = mt * 16 + v + (lanehi << 3);
    outp[(size_t)(gb0 + row) * Dm + n] = fmaxf(acc[v] + b2L[n], 0.f);
  }
}

// ========================= launcher =========================
extern "C" void kernel_launch(void* const* d_in, const int* in_sizes, int n_in,
                              void* d_out, int out_size, void* d_ws, size_t ws_size,
                              hipStream_t stream) {
  (void)in_sizes; (void)n_in; (void)out_size; (void)ws_size;
  const int*   user       = (const int*)d_in[0];
  const int*   cand       = (const int*)d_in[1];
  const int*   hist_idx   = (const int*)d_in[2];
  const int*   hist_r     = (const int*)d_in[3];
  const int*   friend_idx = (const int*)d_in[4];
  const float* user_emb   = (const float*)d_in[5];
  const float* item_emb   = (const float*)d_in[6];
  const float* r2e        = (const float*)d_in[7];
  const float* ratt       = (const float*)d_in[8];
  const float* transform  = (const float*)d_in[9];
  const float* W_lin      = (const float*)d_in[10];
  const float* b_lin      = (const float*)d_in[11];
  const float* W_lin2     = (const float*)d_in[12];
  const float* b_lin2     = (const float*)d_in[13];
  const float* W_att1     = (const float*)d_in[14];
  const float* b_att1     = (const float*)d_in[15];
  const float* W_att5     = (const float*)d_in[16];
  const float* b_att5     = (const float*)d_in[17];

  float* ws     = (float*)d_ws;              // needs 3 * B * D * 4 = 3 MB
  float* queryW = ws;
  float* agg1   = ws + (size_t)Bsz * Dm;
  float* agg2   = ws + 2 * (size_t)Bsz * Dm;

  k_query <<<Bsz / 32, 256, 0, stream>>>(user, cand, user_emb, item_emb,
                                         W_lin, b_lin, queryW);
  k_hist  <<<Bsz,      256, 0, stream>>>(hist_idx, hist_r, item_emb, r2e,
                                         ratt, queryW, agg1);
  k_friend<<<Bsz,      256, 0, stream>>>(user, friend_idx, user_emb, transform,
                                         W_att1, b_att1, queryW, agg2);
  k_final <<<Bsz / 32, 256, 0, stream>>>(user, user_emb, agg1, agg2,
                                         W_att5, b_att5, W_lin2, b_lin2,
                                         (float*)d_out);
}